// GATPolicy_18081812316679
// MI455X (gfx1250) — compile-verified
//
#include <hip/hip_runtime.h>
#include <hip/hip_bf16.h>

typedef float v2f __attribute__((ext_vector_type(2)));
typedef float v8f __attribute__((ext_vector_type(8)));

#define FOUT 256        // HEADS*HID
#define HEADS 4
#define HID 64
#define NEG_SLOPE 0.2f
#define EPS 1e-16f

// ---------- helpers ----------
__device__ __forceinline__ unsigned f2ord(float f) {
  unsigned u = __float_as_uint(f);
  return (u & 0x80000000u) ? ~u : (u | 0x80000000u);
}
__device__ __forceinline__ float ord2f(unsigned u) {
  return (u & 0x80000000u) ? __uint_as_float(u & 0x7fffffffu) : __uint_as_float(~u);
}

__global__ void fill_f32(float* __restrict__ p, float v, int n) {
  int i = blockIdx.x * blockDim.x + threadIdx.x;
  if (i < n) p[i] = v;
}

// ---------- fp32 WMMA GEMM: H[M,256] = X[M,K] @ W[K,256] ----------
// One wave -> one 16x16 output tile. K stepped by 4 (V_WMMA_F32_16X16X4_F32).
// A 16x4 layout: lanes 0-15 hold row M=lane, K={k,k+1}; lanes 16-31 row M=lane-16, K={k+2,k+3}.
// B 4x16 layout: lanes 0-15 hold col N=lane, K={k,k+1}; lanes 16-31 col N=lane-16, K={k+2,k+3}.
// C/D: vgpr r -> (M = r + 8*half, N = lane&15).
__global__ void gemm_wmma_f32(const float* __restrict__ X, const float* __restrict__ W,
                              float* __restrict__ H, int M, int K) {
  int wave = (int)((blockIdx.x * blockDim.x + threadIdx.x) >> 5);
  int lane = threadIdx.x & 31;
  int mt = wave >> 4;           // 16 col-tiles (256/16)
  int nt = wave & 15;
  if (mt * 16 >= M) return;     // uniform across wave
  int half = lane >> 4;
  int l = lane & 15;
  int m0 = mt * 16, n0 = nt * 16;

  const float* __restrict__ xrow = X + (size_t)(m0 + l) * K;
  v8f c = {};
  for (int k = 0; k < K; k += 4) {
    int kk = k + half * 2;
    v2f a;
    a.x = xrow[kk];
    a.y = xrow[kk + 1];
    v2f b;
    b.x = W[(size_t)kk * FOUT + n0 + l];
    b.y = W[(size_t)(kk + 1) * FOUT + n0 + l];
    c = __builtin_amdgcn_wmma_f32_16x16x4_f32(false, a, false, b, (short)0, c,
                                              false, false);
  }
  float* __restrict__ out = H + (size_t)(m0 + half * 8) * FOUT + n0 + l;
#pragma unroll
  for (int r = 0; r < 8; ++r) out[(size_t)r * FOUT] = c[r];
}

// ---------- per-(node,head) attention dots: one wave each ----------
__global__ void node_attn(const float* __restrict__ H, const float* __restrict__ att_src,
                          const float* __restrict__ att_dst, float* __restrict__ a_src,
                          float* __restrict__ a_dst, int N) {
  int wave = (int)((blockIdx.x * blockDim.x + threadIdx.x) >> 5);
  int lane = threadIdx.x & 31;
  int n = wave >> 2, hh = wave & 3;
  if (n >= N) return;
  const float2 hv = *(const float2*)(H + (size_t)n * FOUT + hh * HID + lane * 2);
  const float2 sv = *(const float2*)(att_src + hh * HID + lane * 2);
  const float2 dv = *(const float2*)(att_dst + hh * HID + lane * 2);
  float s = hv.x * sv.x + hv.y * sv.y;
  float d = hv.x * dv.x + hv.y * dv.y;
#pragma unroll
  for (int off = 16; off; off >>= 1) {
    s += __shfl_xor(s, off, 32);
    d += __shfl_xor(d, off, 32);
  }
  if (lane == 0) {
    a_src[n * 4 + hh] = s;
    a_dst[n * 4 + hh] = d;
  }
}

// ---------- fold We @ att_edge -> ve[16][4] (kills the E x 256 GEMM) ----------
__global__ void make_ve(const float* __restrict__ We, const float* __restrict__ att_edge,
                        float* __restrict__ ve) {
  int t = threadIdx.x;  // 64 threads
  if (t >= 64) return;
  int d = t >> 2, hh = t & 3;
  float s = 0.f;
#pragma unroll 8
  for (int cc = 0; cc < HID; ++cc) s += We[d * FOUT + hh * HID + cc] * att_edge[hh * HID + cc];
  ve[d * 4 + hh] = s;
}

// ---------- edge pass 1: logits + segment max (ordered-uint atomicMax) ----------
__global__ void edge_logits(const float* __restrict__ edge_attr, const int* __restrict__ src,
                            const int* __restrict__ dst, const float* __restrict__ a_src,
                            const float* __restrict__ a_dst, const float* __restrict__ ve,
                            float* __restrict__ alpha, unsigned* __restrict__ mord, int E) {
  int e = blockIdx.x * blockDim.x + threadIdx.x;
  if (e >= E) return;
  int s = src[e], d = dst[e];
  float ae0 = 0.f, ae1 = 0.f, ae2 = 0.f, ae3 = 0.f;
  const float4* ap = (const float4*)(edge_attr + (size_t)e * 16);
#pragma unroll
  for (int q = 0; q < 4; ++q) {
    float4 v = ap[q];
    const float* vp = ve + q * 16;
    ae0 += v.x * vp[0] + v.y * vp[4] + v.z * vp[8]  + v.w * vp[12];
    ae1 += v.x * vp[1] + v.y * vp[5] + v.z * vp[9]  + v.w * vp[13];
    ae2 += v.x * vp[2] + v.y * vp[6] + v.z * vp[10] + v.w * vp[14];
    ae3 += v.x * vp[3] + v.y * vp[7] + v.z * vp[11] + v.w * vp[15];
  }
  float4 asv = *(const float4*)(a_src + (size_t)s * 4);
  float4 adv = *(const float4*)(a_dst + (size_t)d * 4);
  float al[4];
  al[0] = asv.x + adv.x + ae0;
  al[1] = asv.y + adv.y + ae1;
  al[2] = asv.z + adv.z + ae2;
  al[3] = asv.w + adv.w + ae3;
#pragma unroll
  for (int hh = 0; hh < 4; ++hh) {
    float a = al[hh];
    a = (a >= 0.f) ? a : NEG_SLOPE * a;
    alpha[(size_t)e * 4 + hh] = a;
    atomicMax(&mord[(size_t)d * 4 + hh], f2ord(a));
  }
}

// ---------- edge pass 2: ea = exp(alpha - m[dst]); denom += ea ----------
__global__ void edge_exp(const int* __restrict__ dst, const unsigned* __restrict__ mord,
                         float* __restrict__ alpha, float* __restrict__ denom, int E) {
  int e = blockIdx.x * blockDim.x + threadIdx.x;
  if (e >= E) return;
  int d = dst[e];
#pragma unroll
  for (int hh = 0; hh < 4; ++hh) {
    float m = ord2f(mord[(size_t)d * 4 + hh]);
    float ea = __expf(alpha[(size_t)e * 4 + hh] - m);
    alpha[(size_t)e * 4 + hh] = ea;  // reuse buffer for ea
    atomicAdd(&denom[(size_t)d * 4 + hh], ea);
  }
}

// ---------- edge pass 3: one wave per edge, gather h[src], scatter-add to acc[dst] ----------
__global__ void edge_message(const int* __restrict__ src, const int* __restrict__ dst,
                             const float* __restrict__ H, const float* __restrict__ ea,
                             const float* __restrict__ denom, float* __restrict__ acc, int E) {
  int wave = (int)((blockIdx.x * blockDim.x + threadIdx.x) >> 5);
  int lane = threadIdx.x & 31;
  if (wave >= E) return;
  int s = src[wave], d = dst[wave];
  int hh = lane >> 3;  // 8 consecutive floats per lane stay within one head
  float w = ea[(size_t)wave * 4 + hh] / (denom[(size_t)d * 4 + hh] + EPS);
  const float4* hp = (const float4*)(H + (size_t)s * FOUT + lane * 8);
  float4 v0 = hp[0], v1 = hp[1];
  float* op = acc + (size_t)d * FOUT + lane * 8;
  atomicAdd(op + 0, v0.x * w);
  atomicAdd(op + 1, v0.y * w);
  atomicAdd(op + 2, v0.z * w);
  atomicAdd(op + 3, v0.w * w);
  atomicAdd(op + 4, v1.x * w);
  atomicAdd(op + 5, v1.y * w);
  atomicAdd(op + 6, v1.z * w);
  atomicAdd(op + 7, v1.w * w);
}

// ---------- bias + relu ----------
__global__ void bias_relu(const float* __restrict__ acc, const float* __restrict__ b,
                          float* __restrict__ xout, int total) {
  int i = blockIdx.x * blockDim.x + threadIdx.x;
  if (i >= total) return;
  float v = acc[i] + b[i & (FOUT - 1)];
  xout[i] = v > 0.f ? v : 0.f;
}

// ---------- global mean pool (wave per node) + tanh ----------
__global__ void pool_accum(const float* __restrict__ X, const int* __restrict__ batch,
                           float* __restrict__ sums, float* __restrict__ counts, int N) {
  int wave = (int)((blockIdx.x * blockDim.x + threadIdx.x) >> 5);
  int lane = threadIdx.x & 31;
  if (wave >= N) return;
  int g = batch[wave];
  const float4* xp = (const float4*)(X + (size_t)wave * FOUT + lane * 8);
  float4 v0 = xp[0], v1 = xp[1];
  float* sp = sums + (size_t)g * FOUT + lane * 8;
  atomicAdd(sp + 0, v0.x);
  atomicAdd(sp + 1, v0.y);
  atomicAdd(sp + 2, v0.z);
  atomicAdd(sp + 3, v0.w);
  atomicAdd(sp + 4, v1.x);
  atomicAdd(sp + 5, v1.y);
  atomicAdd(sp + 6, v1.z);
  atomicAdd(sp + 7, v1.w);
  if (lane == 0) atomicAdd(counts + g, 1.0f);
}

__global__ void pool_final(const float* __restrict__ sums, const float* __restrict__ counts,
                           float* __restrict__ out, int total) {
  int i = blockIdx.x * blockDim.x + threadIdx.x;
  if (i >= total) return;
  float c = counts[i >> 8];
  c = c < 1.f ? 1.f : c;
  out[i] = tanhf(sums[i] / c);
}

// ---------- orchestration ----------
extern "C" void kernel_launch(void* const* d_in, const int* in_sizes, int n_in,
                              void* d_out, int out_size, void* d_ws, size_t ws_size,
                              hipStream_t stream) {
  const float* x         = (const float*)d_in[0];
  const int*   edge_idx  = (const int*)d_in[1];
  const float* edge_attr = (const float*)d_in[2];
  const int*   batch     = (const int*)d_in[3];

  const int N = in_sizes[0] / 64;        // 20000
  const int E = in_sizes[2] / 16;        // 320000
  const int G = 64;
  const int* srcp = edge_idx;
  const int* dstp = edge_idx + E;

  // workspace carve-out (all 16B-aligned slices)
  float* f = (float*)d_ws;
  float*    buf_h  = f; f += (size_t)N * FOUT;
  float*    buf_x  = f; f += (size_t)N * FOUT;
  float*    acc    = f; f += (size_t)N * FOUT;
  float*    a_src  = f; f += (size_t)N * 4;
  float*    a_dst  = f; f += (size_t)N * 4;
  float*    denom  = f; f += (size_t)N * 4;
  unsigned* mord   = (unsigned*)f; f += (size_t)N * 4;
  float*    alpha  = f; f += (size_t)E * 4;   // logits, then ea in-place
  float*    ve     = f; f += 64;
  float*    sums   = f; f += (size_t)G * FOUT;
  float*    counts = f; f += G;

  const int BT = 256;
  auto blocks = [](long long n, int bt) { return (int)((n + bt - 1) / bt); };

  for (int layer = 0; layer < 3; ++layer) {
    const float* W        = (const float*)d_in[4 + layer * 6 + 0];
    const float* We       = (const float*)d_in[4 + layer * 6 + 1];
    const float* att_src  = (const float*)d_in[4 + layer * 6 + 2];
    const float* att_dst  = (const float*)d_in[4 + layer * 6 + 3];
    const float* att_edge = (const float*)d_in[4 + layer * 6 + 4];
    const float* b        = (const float*)d_in[4 + layer * 6 + 5];
    const float* X = (layer == 0) ? x : buf_x;
    const int K = (layer == 0) ? 64 : FOUT;

    // zero accumulators
    fill_f32<<<blocks((long long)N * FOUT, BT), BT, 0, stream>>>(acc, 0.f, N * FOUT);
    fill_f32<<<blocks((long long)N * 4, BT), BT, 0, stream>>>(denom, 0.f, N * 4);
    fill_f32<<<blocks((long long)N * 4, BT), BT, 0, stream>>>((float*)mord, 0.f, N * 4);

    // h = X @ W  (fp32 WMMA)
    long long gemm_waves = (long long)(N / 16) * (FOUT / 16);
    gemm_wmma_f32<<<blocks(gemm_waves * 32, BT), BT, 0, stream>>>(X, W, buf_h, N, K);

    // attention coefficients
    node_attn<<<blocks((long long)N * 4 * 32, BT), BT, 0, stream>>>(buf_h, att_src, att_dst,
                                                                    a_src, a_dst, N);
    make_ve<<<1, 64, 0, stream>>>(We, att_edge, ve);

    // softmax over destination neighborhoods
    edge_logits<<<blocks(E, BT), BT, 0, stream>>>(edge_attr, srcp, dstp, a_src, a_dst, ve,
                                                  alpha, mord, E);
    edge_exp<<<blocks(E, BT), BT, 0, stream>>>(dstp, mord, alpha, denom, E);

    // weighted message aggregation
    edge_message<<<blocks((long long)E * 32, BT), BT, 0, stream>>>(srcp, dstp, buf_h, alpha,
                                                                   denom, acc, E);

    // bias + relu -> input of next layer
    bias_relu<<<blocks((long long)N * FOUT, BT), BT, 0, stream>>>(acc, b, buf_x, N * FOUT);
  }

  // global mean pool + tanh
  fill_f32<<<blocks((long long)G * FOUT, BT), BT, 0, stream>>>(sums, 0.f, G * FOUT);
  fill_f32<<<1, G, 0, stream>>>(counts, 0.f, G);
  pool_accum<<<blocks((long long)N * 32, BT), BT, 0, stream>>>(buf_x, batch, sums, counts, N);
  pool_final<<<blocks((long long)G * FOUT, BT), BT, 0, stream>>>(sums, counts, (float*)d_out,
                                                                 G * FOUT);
}